// MemoryBank_46179488367385
// MI455X (gfx1250) — compile-verified
//
#include <hip/hip_runtime.h>
#include <hip/hip_bf16.h>
#include <math.h>

#define SIZE   131072
#define DIM    128
#define BATCH  1024
#define KNN    16

typedef __attribute__((ext_vector_type(2))) float v2f;
typedef __attribute__((ext_vector_type(4))) float v4f;
typedef __attribute__((ext_vector_type(8))) float v8f;

// ---------------------------------------------------------------------------
// Kernel 1: row sums-of-squares for emb (1024 rows) and bank (131072 rows).
// One wave per row: 32 lanes x float4 = 128 floats, then shfl_xor reduce.
// ---------------------------------------------------------------------------
__global__ void rowsq_kernel(const float* __restrict__ emb,
                             const float* __restrict__ bank,
                             float* __restrict__ e_sq,
                             float* __restrict__ b_sq) {
    const int wave = (int)((blockIdx.x * blockDim.x + threadIdx.x) >> 5);
    const int lane = (int)(threadIdx.x & 31);
    const float* src;
    float* dst;
    int row;
    if (wave < BATCH) {
        src = emb; dst = e_sq; row = wave;
    } else {
        row = wave - BATCH;
        if (row >= SIZE) return;
        src = bank; dst = b_sq;
    }
    const v4f* p = (const v4f*)(src + (size_t)row * DIM);
    v4f v = p[lane];
    float s = v.x * v.x + v.y * v.y + v.z * v.z + v.w * v.w;
#pragma unroll
    for (int off = 16; off > 0; off >>= 1) s += __shfl_xor(s, off, 32);
    if (lane == 0) dst[row] = s;
}

// ---------------------------------------------------------------------------
// Kernel 2: fused GEMM (all_dots) + distances via V_WMMA_F32_16X16X4_F32.
// Block = 256 threads (8 waves). Wave w computes a 16(M) x 64(N) strip:
// A-frags for full K=128 held in registers, 4 N-subtiles of 16x16.
// ---------------------------------------------------------------------------
__global__ void __launch_bounds__(256)
dots_kernel(const float* __restrict__ emb,
            const float* __restrict__ bank,
            const float* __restrict__ e_sq,
            const float* __restrict__ b_sq,
            float* __restrict__ dist_out,
            float* __restrict__ dots_out) {
    const int lane   = (int)(threadIdx.x & 31);
    const int wave   = (int)(threadIdx.x >> 5);            // 0..7
    const int m_base = (int)blockIdx.y * 16;               // emb row tile
    const int s_base = (int)blockIdx.x * 512 + wave * 64;  // bank row tile

    const int r15   = lane & 15;
    const int khalf = (lane >> 4) * 2;  // lanes 0-15 -> K=+0,+1 ; 16-31 -> K=+2,+3

    // A fragments: 32 k-steps of 16x4 covering K=128 (64 VGPRs, reused 4x)
    v2f a[32];
    const float* arow = emb + (size_t)(m_base + r15) * DIM + khalf;
#pragma unroll
    for (int kk = 0; kk < 32; ++kk)
        a[kk] = *(const v2f*)(arow + kk * 4);

    // e_sq for the 8 output rows this lane-half touches
    float es[8];
    const int mrow = m_base + (lane >> 4) * 8;
#pragma unroll
    for (int r = 0; r < 8; ++r) es[r] = e_sq[mrow + r];

#pragma unroll
    for (int t = 0; t < 4; ++t) {
        const int s0 = s_base + t * 16;
        const float* brow = bank + (size_t)(s0 + r15) * DIM + khalf;

        v8f acc = {};
#pragma unroll
        for (int kk = 0; kk < 32; ++kk) {
            v2f b = *(const v2f*)(brow + kk * 4);
            acc = __builtin_amdgcn_wmma_f32_16x16x4_f32(
                /*neg_a=*/false, a[kk], /*neg_b=*/false, b,
                /*c_mod=*/(short)0, acc, /*reuse_a=*/false, /*reuse_b=*/false);
        }

        const float bs  = b_sq[s0 + r15];
        const size_t col = (size_t)(s0 + r15);
#pragma unroll
        for (int r = 0; r < 8; ++r) {
            const size_t off = (size_t)(mrow + r) * SIZE + col;
            const float d = acc[r];
            __builtin_nontemporal_store(d, dots_out + off);
            float dd = es[r] + bs - 2.0f * d;
            dd = dd > 0.0f ? sqrtf(dd) : 0.0f;
            __builtin_nontemporal_store(dd, dist_out + off);
        }
    }
}

// ---------------------------------------------------------------------------
// Kernel 3: knn_dots[b,k] = dot(bank[idxs[b,k]], emb[b]). One wave per (b,k).
// ---------------------------------------------------------------------------
__global__ void knn_kernel(const float* __restrict__ emb,
                           const float* __restrict__ bank,
                           const int* __restrict__ idxs,
                           float* __restrict__ out) {
    const int gw   = (int)((blockIdx.x * blockDim.x + threadIdx.x) >> 5);
    const int lane = (int)(threadIdx.x & 31);
    if (gw >= BATCH * KNN) return;
    const int b   = gw >> 4;           // gw / KNN
    const int row = idxs[gw];
    const v4f* e = (const v4f*)(emb + (size_t)b * DIM);
    const v4f* g = (const v4f*)(bank + (size_t)row * DIM);
    v4f ev = e[lane], gv = g[lane];
    float s = ev.x * gv.x + ev.y * gv.y + ev.z * gv.z + ev.w * gv.w;
#pragma unroll
    for (int off = 16; off > 0; off >>= 1) s += __shfl_xor(s, off, 32);
    if (lane == 0) out[gw] = s;
}

// ---------------------------------------------------------------------------
// Kernel 4: copy bank -> new_bank (float4, nontemporal stores).
// ---------------------------------------------------------------------------
__global__ void copy_kernel(const v4f* __restrict__ src, v4f* __restrict__ dst) {
    const size_t i = (size_t)blockIdx.x * blockDim.x + threadIdx.x;
    __builtin_nontemporal_store(src[i], dst + i);
}

// ---------------------------------------------------------------------------
// Kernel 5: scatter data_memory rows into new_bank at update_indices.
// ---------------------------------------------------------------------------
__global__ void scatter_kernel(const float* __restrict__ data_memory,
                               const int* __restrict__ upd,
                               float* __restrict__ new_bank) {
    const int b = (int)blockIdx.x;   // 0..BATCH-1
    const int t = (int)threadIdx.x;  // 0..DIM-1
    const int row = upd[b];
    new_bank[(size_t)row * DIM + t] = data_memory[(size_t)b * DIM + t];
}

// ---------------------------------------------------------------------------
extern "C" void kernel_launch(void* const* d_in, const int* in_sizes, int n_in,
                              void* d_out, int out_size, void* d_ws, size_t ws_size,
                              hipStream_t stream) {
    const float* emb   = (const float*)d_in[0];  // [BATCH, DIM]
    const float* dmem  = (const float*)d_in[1];  // [BATCH, DIM]
    const float* bank  = (const float*)d_in[2];  // [SIZE, DIM]
    const int*   idxs  = (const int*)d_in[3];    // [BATCH, KNN]
    const int*   upd   = (const int*)d_in[4];    // [BATCH]

    float* out = (float*)d_out;
    const size_t bs_elems = (size_t)BATCH * SIZE;
    float* dist_out = out;                                   // [B,S]
    float* dots_out = out + bs_elems;                        // [B,S]
    float* knn_out  = out + 2 * bs_elems;                    // [B,K]
    float* bank_out = out + 2 * bs_elems + BATCH * KNN;      // [S,D]

    float* e_sq = (float*)d_ws;            // 1024 floats
    float* b_sq = e_sq + BATCH;            // 131072 floats

    // 1) row norms
    {
        const int waves = BATCH + SIZE;          // 132096
        const int blocks = waves / 8;            // 256 threads = 8 waves
        rowsq_kernel<<<blocks, 256, 0, stream>>>(emb, bank, e_sq, b_sq);
    }
    // 2) fused WMMA GEMM + distances
    {
        dim3 grid(SIZE / 512, BATCH / 16);       // (256, 64)
        dots_kernel<<<grid, 256, 0, stream>>>(emb, bank, e_sq, b_sq,
                                              dist_out, dots_out);
    }
    // 3) knn dots
    {
        const int waves = BATCH * KNN;           // 16384
        knn_kernel<<<waves / 8, 256, 0, stream>>>(emb, bank, idxs, knn_out);
    }
    // 4) bank copy
    {
        const size_t n4 = (size_t)SIZE * DIM / 4;   // 4,194,304
        copy_kernel<<<(unsigned)(n4 / 256), 256, 0, stream>>>((const v4f*)bank,
                                                              (v4f*)bank_out);
    }
    // 5) scatter updates (after copy; stream-ordered)
    scatter_kernel<<<BATCH, DIM, 0, stream>>>(dmem, upd, bank_out);
}